// CRFLoss_74474732913014
// MI455X (gfx1250) — compile-verified
//
#include <hip/hip_runtime.h>
#include <stdint.h>

#define NEGV  (-10000.0f)
#define LN2F  (0.69314718055994531f)
#define ILN2F (1.44269504088896340f)

// Base-2 logsumexp helpers: bare v_exp_f32 / v_log_f32, no hidden muls.
__device__ __forceinline__ float lse2_2(float a, float b) {
    float m = fmaxf(a, b);
    return m + __builtin_amdgcn_logf(__builtin_amdgcn_exp2f(a - m) +
                                     __builtin_amdgcn_exp2f(b - m));
}
__device__ __forceinline__ float lse3_2(float a, float b, float c) {
    float m = fmaxf(fmaxf(a, b), c);
    return m + __builtin_amdgcn_logf(__builtin_amdgcn_exp2f(a - m) +
                                     __builtin_amdgcn_exp2f(b - m) +
                                     __builtin_amdgcn_exp2f(c - m));
}
__device__ __forceinline__ float lse4_2(float a, float b, float c, float d) {
    float m = fmaxf(fmaxf(a, b), fmaxf(c, d));
    return m + __builtin_amdgcn_logf(__builtin_amdgcn_exp2f(a - m) +
                                     __builtin_amdgcn_exp2f(b - m) +
                                     __builtin_amdgcn_exp2f(c - m) +
                                     __builtin_amdgcn_exp2f(d - m));
}

// ---------------------------------------------------------------------------
// Alignment (path) score: per batch, sum of gathered emissions + transitions.
// (Natural domain: pure sums, no logs.)
// ---------------------------------------------------------------------------
__global__ __launch_bounds__(256) void crf_ali_kernel(
    const float* __restrict__ obs, const float* __restrict__ trans,
    const float* __restrict__ P, const int* __restrict__ ali,
    float* __restrict__ ali_out)
{
    constexpr int Y = 384, L = 512;
    const int b = blockIdx.x;
    const int tid = threadIdx.x;
    const int* a = ali + (size_t)b * L * 3;
    const float* obs_b = obs + (size_t)b * 384 * 384 * 3;

    float acc = 0.0f;
    for (int t = tid; t < L; t += 256) {
        int x = a[t * 3 + 0];
        int y = a[t * 3 + 1];
        int s = a[t * 3 + 2];
        acc += obs_b[((size_t)(x - 1) * Y + (y - 1)) * 3 + s];
        if (t < L - 1) {
            int s2 = a[(t + 1) * 3 + 2];
            acc += trans[s * 5 + s2] + P[s * 5 + s2];
        }
    }
    __shared__ float red[256];
    red[tid] = acc;
    __syncthreads();
    for (int s = 128; s > 0; s >>= 1) {
        if (tid < s) red[tid] += red[tid + s];
        __syncthreads();
    }
    if (tid == 0) ali_out[b] = red[0];
}

// ---------------------------------------------------------------------------
// CRF partition via anti-diagonal wavefront DP, entirely in log2 domain.
// One workgroup per batch; thread tid owns row i = tid+1 of each diagonal.
// Rolling 3-diagonal DP state in LDS. Emissions prefetched 4 diagonals ahead
// through a 5-slot async global->LDS pipeline (ASYNCcnt), issued at the top
// of each step into the slot consumed last step, synced with s_wait_asynccnt.
// ---------------------------------------------------------------------------
__global__ __launch_bounds__(384) void crf_dp_kernel(
    const float* __restrict__ obs, const float* __restrict__ trans,
    const float* __restrict__ P, const int* __restrict__ maskX,
    const int* __restrict__ maskY, const float* __restrict__ ali_ws,
    float* __restrict__ out)
{
    constexpr int X = 384, Y = 384, DMAX = X + Y; // diagonals d = 2..768
    const int b = blockIdx.x;
    const int tid = threadIdx.x;
    const int i = tid + 1; // 1..384

    __shared__ float dp[3][3][X + 1];   // [diag%3][state][i]  (log2 units)
    __shared__ float ebuf[5][X * 3];    // 5-slot emission pipeline (raw f32)
    __shared__ float redM[384];
    __shared__ float redS[384];

    // Effective transitions T = (trans + P) scaled to log2 units.
    const float T00 = (trans[0] + P[0]) * ILN2F;
    const float T01 = (trans[1] + P[1]) * ILN2F;
    const float T02 = (trans[2] + P[2]) * ILN2F;
    const float T10 = (trans[5] + P[5]) * ILN2F;
    const float T11 = (trans[6] + P[6]) * ILN2F;
    const float T12 = (trans[7] + P[7]) * ILN2F;
    const float T20 = (trans[10] + P[10]) * ILN2F;
    const float T22 = (trans[12] + P[12]) * ILN2F;
    // trans[11] (Y->X) is structurally NEG: its exp underflows to exactly 0
    // whenever any sibling term is live -> dropped from the X-state lse.
    const float thm = (trans[15] + P[15]) * ILN2F; // head -> M
    const float tmt = (trans[4]  + P[4])  * ILN2F; // M -> tail
    const int mX = maskX[b];
    const int mY = maskY[b];

    // Init rolling DP buffers to NEG (log2 units; magnitude only gates exps).
    {
        float* dpf = &dp[0][0][0];
        for (int k = tid; k < 3 * 3 * (X + 1); k += 384) dpf[k] = NEGV * ILN2F;
    }
    __syncthreads();

    const float* obs_b = obs + (size_t)b * X * Y * 3;
    const uint32_t ldsE = (uint32_t)(uintptr_t)(&ebuf[0][0]);

    // Issue 3 async dword loads for this thread's cell on diagonal d_target
    // into pipeline slot `slot`. Addresses clamped in-bounds and issued
    // unconditionally: every wave issues exactly 3 async instructions per call
    // (exact ASYNCcnt bookkeeping, no EXEC-dependent skips). The instruction
    // offset is added to BOTH the global and LDS address, so one base covers
    // the 3 consecutive floats of the cell.
    auto issue = [&](int d_target, int slot) {
        int dn = d_target > DMAX ? DMAX : d_target;
        int jn = dn - i;
        jn = jn < 1 ? 1 : (jn > Y ? Y : jn);
        uint32_t off = (uint32_t)((tid * Y + (jn - 1)) * 3) * 4u;
        uint64_t ga = (uint64_t)(uintptr_t)obs_b + off;
        uint32_t dst = ldsE + ((uint32_t)slot * (uint32_t)(X * 3) +
                               (uint32_t)tid * 3u) * 4u;
        asm volatile(
            "global_load_async_to_lds_b32 %0, %1, off\n\t"
            "global_load_async_to_lds_b32 %0, %1, off offset:4\n\t"
            "global_load_async_to_lds_b32 %0, %1, off offset:8"
            :: "v"(dst), "v"(ga) : "memory");
    };

    // Prologue: fill pipeline with diagonals 2..5 (slots 2,3,4,0).
    issue(2, 2); issue(3, 3); issue(4, 4); issue(5, 0);

    // Streaming base-2 logsumexp accumulator (running max + sum of exp2).
    float zM = -1e30f;
    float zS = 0.0f;

    int cur = 2, p1 = 1, p2 = 0;   // d%3, (d-1)%3, (d-2)%3 for d=2
    int ebr = 2, ebw = 1;          // read slot d%5, write slot (d+4)%5 for d=2

    for (int d = 2; d <= DMAX; ++d) {
        // Refill the slot consumed last step with diagonal d+4, *before* this
        // step's reads/compute, so the loads overlap this step and the next 3.
        issue(d + 4, ebw);
        // Up to 15 async ops outstanding (d..d+4); waiting to <=12 completes
        // the oldest 3, i.e. exactly this diagonal's triple (in-order returns).
        asm volatile("s_wait_asynccnt 12" ::: "memory");

        const int j = d - i;
        const bool valid = (j >= 1) && (j <= Y);

        const float e0 = ebuf[ebr][tid * 3 + 0] * ILN2F;
        const float e1 = ebuf[ebr][tid * 3 + 1] * ILN2F;
        const float e2 = ebuf[ebr][tid * 3 + 2] * ILN2F;

        const float pM0 = dp[p2][0][i - 1], pM1 = dp[p2][1][i - 1], pM2 = dp[p2][2][i - 1];
        const float pX0 = dp[p1][0][i - 1], pX1 = dp[p1][1][i - 1];
        const float pY0 = dp[p1][0][i],     pY1 = dp[p1][1][i],     pY2 = dp[p1][2][i];

        float m = e0 + lse4_2(pM0 + T00, pM1 + T10, pM2 + T20, thm);
        float x = e1 + lse2_2(pX0 + T01, pX1 + T11);
        float y = e2 + lse3_2(pY0 + T02, pY1 + T12, pY2 + T22);
        if (!valid) { m = NEGV * ILN2F; x = NEGV * ILN2F; y = NEGV * ILN2F; }

        dp[cur][0][i] = m;
        dp[cur][1][i] = x;
        dp[cur][2][i] = y;

        if (valid && i <= mX && j <= mY) {
            float c = m + tmt;
            float nM = fmaxf(zM, c);
            zS = zS * __builtin_amdgcn_exp2f(zM - nM) +
                 __builtin_amdgcn_exp2f(c - nM);
            zM = nM;
        }

        // Rotate DP buffers (mod 3) and pipeline slots (mod 5).
        int t3 = p2; p2 = p1; p1 = cur; cur = t3;
        ebr = (ebr == 4) ? 0 : ebr + 1;
        ebw = (ebw == 4) ? 0 : ebw + 1;

        __syncthreads();
    }

    // Block-wide base-2 logsumexp reduction of (zM, zS) over 384 threads.
    redM[tid] = zM;
    redS[tid] = zS;
    __syncthreads();
    if (tid < 128) {
        float M1 = redM[tid], S1 = redS[tid];
        {
            float M2 = redM[tid + 128], S2 = redS[tid + 128];
            float M = fmaxf(M1, M2);
            S1 = S1 * __builtin_amdgcn_exp2f(M1 - M) +
                 S2 * __builtin_amdgcn_exp2f(M2 - M);
            M1 = M;
        }
        {
            float M2 = redM[tid + 256], S2 = redS[tid + 256];
            float M = fmaxf(M1, M2);
            S1 = S1 * __builtin_amdgcn_exp2f(M1 - M) +
                 S2 * __builtin_amdgcn_exp2f(M2 - M);
            M1 = M;
        }
        redM[tid] = M1;
        redS[tid] = S1;
    }
    __syncthreads();
    for (int s = 64; s > 0; s >>= 1) {
        if (tid < s) {
            float M1 = redM[tid], S1 = redS[tid];
            float M2 = redM[tid + s], S2 = redS[tid + s];
            float M = fmaxf(M1, M2);
            redS[tid] = S1 * __builtin_amdgcn_exp2f(M1 - M) +
                        S2 * __builtin_amdgcn_exp2f(M2 - M);
            redM[tid] = M;
        }
        __syncthreads();
    }

    if (tid == 0) {
        // Convert log2 -> natural log once, at the very end.
        float Z = LN2F * (redM[0] + __builtin_amdgcn_logf(redS[0]));
        out[b] = Z - ali_ws[b];
    }
}

// ---------------------------------------------------------------------------
extern "C" void kernel_launch(void* const* d_in, const int* in_sizes, int n_in,
                              void* d_out, int out_size, void* d_ws, size_t ws_size,
                              hipStream_t stream) {
    const float* obs        = (const float*)d_in[0]; // (64,384,384,3) f32
    const float* trans      = (const float*)d_in[1]; // (5,5) f32
    const float* P          = (const float*)d_in[2]; // (5,5) f32
    const int*   alignments = (const int*)d_in[3];   // (64,512,3) int
    const int*   maskX      = (const int*)d_in[4];   // (64,) int
    const int*   maskY      = (const int*)d_in[5];   // (64,) int
    float* out    = (float*)d_out;                   // (64,) f32
    float* ali_ws = (float*)d_ws;                    // 64 floats of scratch

    crf_ali_kernel<<<64, 256, 0, stream>>>(obs, trans, P, alignments, ali_ws);
    crf_dp_kernel<<<64, 384, 0, stream>>>(obs, trans, P, maskX, maskY, ali_ws, out);
}